// SVDRegistration_80573586473569
// MI455X (gfx1250) — compile-verified
//
#include <hip/hip_runtime.h>
#include <hip/hip_bf16.h>

typedef float v2f __attribute__((ext_vector_type(2)));
typedef float v8f __attribute__((ext_vector_type(8)));

#define BATCHES 4096
#define NPTS    2048

// ---------------------------------------------------------------------------
// Phase 1: per-batch moment reduction via V_WMMA_F32_16X16X4_F32.
// One wave (32 threads) handles 4 batches. The 16x16 accumulator tile holds
// four 4x4 diagonal blocks, each = sum_n [w*t_x, w*t_y, w*t_z, w]^T (x)
// [s_x, s_y, s_z, 1]  ==> all 16 moments needed for the weighted Procrustes.
//
// LDS layout: SoA slabs A[row][point], B[row][point] with row = 4*bi + comp,
// padded to 34 floats per row. A lane's two K-points are consecutive floats
// at an even dword offset -> one 8B-aligned ds_load_b64 straight into the
// WMMA source register pair (no deinterleave movs).
// ---------------------------------------------------------------------------
__global__ __launch_bounds__(32)
void svdreg_moments_wmma(const float* __restrict__ target,
                         const float* __restrict__ source,
                         const float* __restrict__ weights,
                         float* __restrict__ moments /* [B/4][64] */)
{
    __shared__ float2 sA[16 * 17];   // 16 rows x 34 floats
    __shared__ float2 sB[16 * 17];
    float* const fA = (float*)sA;
    float* const fB = (float*)sB;

    const int lane = threadIdx.x;          // 0..31
    const int tile = blockIdx.x;           // 0..1023
    const int b0   = tile * 4;

    // WMMA operand indexing for this lane (A row m and B column n are both
    // lane&15, so the LDS index math is shared).
    const int m      = lane & 15;
    const int g      = lane >> 4;          // K-half: VGPR0/1 hold K=2g, 2g+1
    const int bi     = m >> 2;             // batch-in-tile of this row/col
    const int comp   = m & 3;              // component 0..3
    const int rowoff = m * 17;             // float2 units: row m slab

    v8f acc{};                             // 16x16 f32 accumulator (zero)

    for (int n0 = 0; n0 < NPTS; n0 += 32) {
        // ---- stage: coalesced NT load of 32 points for each of 4 batches --
        #pragma unroll
        for (int q = 0; q < 4; ++q) {
            const size_t pidx = (size_t)(b0 + q) * NPTS + n0 + lane;
            const float* tp = target + pidx * 3;
            const float* sp = source + pidx * 3;
            // prefetch next chunk of this batch's stream
            __builtin_prefetch(tp + 32 * 3, 0, 3);
            __builtin_prefetch(sp + 32 * 3, 0, 3);
            const float w  = __builtin_nontemporal_load(weights + pidx);
            const float t0 = __builtin_nontemporal_load(tp + 0);
            const float t1 = __builtin_nontemporal_load(tp + 1);
            const float t2 = __builtin_nontemporal_load(tp + 2);
            const float s0 = __builtin_nontemporal_load(sp + 0);
            const float s1 = __builtin_nontemporal_load(sp + 1);
            const float s2 = __builtin_nontemporal_load(sp + 2);
            const int r = q * 4;           // base row of this batch
            fA[(r + 0) * 34 + lane] = w * t0;
            fA[(r + 1) * 34 + lane] = w * t1;
            fA[(r + 2) * 34 + lane] = w * t2;
            fA[(r + 3) * 34 + lane] = w;
            fB[(r + 0) * 34 + lane] = s0;
            fB[(r + 1) * 34 + lane] = s1;
            fB[(r + 2) * 34 + lane] = s2;
            fB[(r + 3) * 34 + lane] = 1.0f;
        }
        // single wave: LDS ops are in-order per wave; no barrier needed.

        // ---- 8 WMMAs consume the 32 staged points (K=4 points each) ----
        #pragma unroll
        for (int c = 0; c < 8; ++c) {
            // this lane's K = 2g, 2g+1 of chunk c: float2 index c*2 + g
            const v2f a = *reinterpret_cast<const v2f*>(&sA[rowoff + c * 2 + g]);
            const v2f b = *reinterpret_cast<const v2f*>(&sB[rowoff + c * 2 + g]);
            acc = __builtin_amdgcn_wmma_f32_16x16x4_f32(
                      false, a, false, b, (short)0, acc, false, false);
        }
    }

    // ---- extract the four useful 4x4 diagonal blocks ----
    // acc element v of this lane is tile element (row M = v + 8g, col N = m).
    const int bj = m >> 2;                 // column's batch-in-tile
    const int j  = m & 3;
    float* outt = moments + (size_t)tile * 64;
    #pragma unroll
    for (int v = 0; v < 8; ++v) {
        const int M = v + 8 * g;
        if (M >= 4 * bj && M < 4 * bj + 4) {
            const int i = M - 4 * bj;
            outt[bj * 16 + i * 4 + j] = acc[v];
        }
    }
    (void)bi; (void)comp;
}

// ---------------------------------------------------------------------------
// Phase 2: one thread per batch. Build cov from raw moments, take the
// orthogonal polar factor (== U*Vh of the SVD, including sign) via Newton
// iteration X <- 0.5*(X + X^-T), then emit the 4x4 transform.
// ---------------------------------------------------------------------------
__global__ __launch_bounds__(256)
void svdreg_solve(const float* __restrict__ moments,
                  float* __restrict__ out /* [B][4][4] */)
{
    const int b = blockIdx.x * blockDim.x + threadIdx.x;
    if (b >= BATCHES) return;

    const float* mm = moments + (size_t)(b >> 2) * 64 + (b & 3) * 16;

    const float W = mm[15];
    float Swt[3], Sws[3], M[3][3];
    #pragma unroll
    for (int i = 0; i < 3; ++i) {
        Swt[i] = mm[i * 4 + 3];
        Sws[i] = mm[12 + i];
        #pragma unroll
        for (int j = 0; j < 3; ++j) M[i][j] = mm[i * 4 + j];
    }

    const float denom = W + 1e-8f;
    float mt[3], ms[3];
    #pragma unroll
    for (int i = 0; i < 3; ++i) { mt[i] = Swt[i] / denom; ms[i] = Sws[i] / denom; }

    // cov_ij = sum w (t_i - mt_i)(s_j - ms_j)
    float X[3][3];
    float fn = 0.0f;
    #pragma unroll
    for (int i = 0; i < 3; ++i)
        #pragma unroll
        for (int j = 0; j < 3; ++j) {
            const float c = M[i][j] - ms[j] * Swt[i] - mt[i] * Sws[j] + mt[i] * ms[j] * W;
            X[i][j] = c;
            fn += c * c;
        }

    // Normalize so singular values are <= 1, then Newton polar iteration.
    const float sc = rsqrtf(fn + 1e-30f);
    #pragma unroll
    for (int i = 0; i < 3; ++i)
        #pragma unroll
        for (int j = 0; j < 3; ++j) X[i][j] *= sc;

    for (int it = 0; it < 16; ++it) {
        const float c00 = X[1][1]*X[2][2] - X[1][2]*X[2][1];
        const float c01 = X[1][2]*X[2][0] - X[1][0]*X[2][2];
        const float c02 = X[1][0]*X[2][1] - X[1][1]*X[2][0];
        const float c10 = X[0][2]*X[2][1] - X[0][1]*X[2][2];
        const float c11 = X[0][0]*X[2][2] - X[0][2]*X[2][0];
        const float c12 = X[0][1]*X[2][0] - X[0][0]*X[2][1];
        const float c20 = X[0][1]*X[1][2] - X[0][2]*X[1][1];
        const float c21 = X[0][2]*X[1][0] - X[0][0]*X[1][2];
        const float c22 = X[0][0]*X[1][1] - X[0][1]*X[1][0];
        const float det = X[0][0]*c00 + X[0][1]*c01 + X[0][2]*c02;
        const float r   = 0.5f / (det + (det >= 0.0f ? 1e-30f : -1e-30f));
        // X^-T = cof(X)/det ; X <- 0.5*(X + X^-T)
        X[0][0] = 0.5f*X[0][0] + c00*r;  X[0][1] = 0.5f*X[0][1] + c01*r;  X[0][2] = 0.5f*X[0][2] + c02*r;
        X[1][0] = 0.5f*X[1][0] + c10*r;  X[1][1] = 0.5f*X[1][1] + c11*r;  X[1][2] = 0.5f*X[1][2] + c12*r;
        X[2][0] = 0.5f*X[2][0] + c20*r;  X[2][1] = 0.5f*X[2][1] + c21*r;  X[2][2] = 0.5f*X[2][2] + c22*r;
    }

    // t = mt - R * ms ; write row-major 4x4
    float* o = out + (size_t)b * 16;
    #pragma unroll
    for (int i = 0; i < 3; ++i) {
        float ti = mt[i];
        #pragma unroll
        for (int j = 0; j < 3; ++j) { o[i * 4 + j] = X[i][j]; ti -= X[i][j] * ms[j]; }
        o[i * 4 + 3] = ti;
    }
    o[12] = 0.0f; o[13] = 0.0f; o[14] = 0.0f; o[15] = 1.0f;
}

extern "C" void kernel_launch(void* const* d_in, const int* in_sizes, int n_in,
                              void* d_out, int out_size, void* d_ws, size_t ws_size,
                              hipStream_t stream)
{
    const float* target  = (const float*)d_in[0];
    const float* source  = (const float*)d_in[1];
    const float* weights = (const float*)d_in[2];
    float* out = (float*)d_out;
    float* mom = (float*)d_ws;   // (BATCHES/4) * 64 floats = 256 KB

    svdreg_moments_wmma<<<BATCHES / 4, 32, 0, stream>>>(target, source, weights, mom);
    svdreg_solve<<<(BATCHES + 255) / 256, 256, 0, stream>>>(mom, out);
}